// FixedPointSolver_18373870092875
// MI455X (gfx1250) — compile-verified
//
#include <hip/hip_runtime.h>
#include <math.h>

typedef __attribute__((ext_vector_type(2))) float v2f;
typedef __attribute__((ext_vector_type(8))) float v8f;

#define D_DIM   1024
#define BATCH   256
#define N_FLAT  (BATCH * D_DIM)   // 262144
#define M_HIST  10
#define NSEG    32
#define MAXPAIR 55

// ---------------------------------------------------------------------------
// f(z) = tanh(z @ W + inj + b) using V_WMMA_F32_16X16X4_F32.
// Block: 256 threads (8 wave32), 128(M) x 64(N) output tile, BK=32 chunks.
// Wave grid 4x2; each wave computes 32x32 = four 16x16 f32 WMMA accumulators
// (2 A-frags + 2 B-frags -> 4 WMMA per K-step: 1.0 LDS load per WMMA).
// LDS: As[128][36]  - A frags are single ds_load_b64, banks conflict-free
//      Bs[16][160]  - K-pair interleaved (Bs[kp][2n+i] = W[2kp+i][n]) so each
//                     B frag is one aligned ds_load_b64; stride 160 floats
//                     puts half-waves 32 banks apart -> conflict-free.
// ---------------------------------------------------------------------------
__global__ __launch_bounds__(256) void f_apply_kernel(
    const float* __restrict__ Z, const float* __restrict__ W,
    const float* __restrict__ inj, const float* __restrict__ bias,
    float* __restrict__ Out)
{
    __shared__ float As[128][36];
    __shared__ float Bs[16][160];

    const int bm = blockIdx.y * 128;
    const int bn = blockIdx.x * 64;
    const int tid   = threadIdx.x;
    const int lane  = tid & 31;
    const int w     = tid >> 5;
    const int wm    = (w >> 1) * 32;   // 0,32,64,96
    const int wn    = (w & 1) * 32;    // 0,32
    const int half  = lane >> 4;       // 0/1
    const int idx16 = lane & 15;

    v8f acc00 = {}, acc01 = {}, acc10 = {}, acc11 = {};

    for (int kk = 0; kk < D_DIM; kk += 32) {
        // Load A tile: 128 rows x 32 cols (Z) = 1024 float4 over 256 threads.
#pragma unroll
        for (int it = 0; it < 4; ++it) {
            int qq  = tid + it * 256;          // 0..1023
            int row = qq >> 3;                 // 8 float4 per row of 32
            int c4  = (qq & 7) * 4;
            float4 v = *reinterpret_cast<const float4*>(Z + (size_t)(bm + row) * D_DIM + kk + c4);
            *reinterpret_cast<float4*>(&As[row][c4]) = v;
        }
        // Load B tile: 32 rows x 64 cols (W), scatter K-pair interleaved.
#pragma unroll
        for (int it = 0; it < 2; ++it) {
            int qq  = tid + it * 256;          // 0..511
            int row = qq >> 4;                 // 16 float4 per row of 64
            int c4  = (qq & 15) * 4;
            float4 v = *reinterpret_cast<const float4*>(W + (size_t)(kk + row) * D_DIM + bn + c4);
            const int kp = row >> 1;
            const int io = row & 1;
            Bs[kp][2 * (c4 + 0) + io] = v.x;
            Bs[kp][2 * (c4 + 1) + io] = v.y;
            Bs[kp][2 * (c4 + 2) + io] = v.z;
            Bs[kp][2 * (c4 + 3) + io] = v.w;
        }
        __syncthreads();

#pragma unroll
        for (int kt = 0; kt < 8; ++kt) {
            const int ka = kt * 4 + 2 * half;  // A: VGPR0 holds K=0 (lanes 0-15) / K=2 (lanes 16-31)
            const int kp = kt * 2 + half;      // B pair row: rows (2kp, 2kp+1)
            v2f a0 = *reinterpret_cast<const v2f*>(&As[wm + idx16][ka]);
            v2f a1 = *reinterpret_cast<const v2f*>(&As[wm + 16 + idx16][ka]);
            v2f b0 = *reinterpret_cast<const v2f*>(&Bs[kp][2 * (wn + idx16)]);
            v2f b1 = *reinterpret_cast<const v2f*>(&Bs[kp][2 * (wn + 16 + idx16)]);
            acc00 = __builtin_amdgcn_wmma_f32_16x16x4_f32(false, a0, false, b0, (short)0, acc00, false, false);
            acc01 = __builtin_amdgcn_wmma_f32_16x16x4_f32(false, a0, false, b1, (short)0, acc01, false, false);
            acc10 = __builtin_amdgcn_wmma_f32_16x16x4_f32(false, a1, false, b0, (short)0, acc10, false, false);
            acc11 = __builtin_amdgcn_wmma_f32_16x16x4_f32(false, a1, false, b1, (short)0, acc11, false, false);
        }
        __syncthreads();
    }

    // Epilogue: C/D layout -> VGPR r: M = r + 8*half, N = idx16.
#pragma unroll
    for (int r = 0; r < 8; ++r) {
        const int gr0 = bm + wm + r + half * 8;        // M-tile 0
        const int gr1 = gr0 + 16;                      // M-tile 1
        const int gc0 = bn + wn + idx16;               // N-tile 0
        const int gc1 = gc0 + 16;                      // N-tile 1
        const float bc0 = bias[gc0];
        const float bc1 = bias[gc1];
        Out[(size_t)gr0 * D_DIM + gc0] = tanhf(acc00[r] + inj[(size_t)gr0 * D_DIM + gc0] + bc0);
        Out[(size_t)gr0 * D_DIM + gc1] = tanhf(acc01[r] + inj[(size_t)gr0 * D_DIM + gc1] + bc1);
        Out[(size_t)gr1 * D_DIM + gc0] = tanhf(acc10[r] + inj[(size_t)gr1 * D_DIM + gc0] + bc0);
        Out[(size_t)gr1 * D_DIM + gc1] = tanhf(acc11[r] + inj[(size_t)gr1 * D_DIM + gc1] + bc1);
    }
}

// ---------------------------------------------------------------------------
// Partial dot products for H = G G^T, G_i = F_i - X_i (valid rows only).
// grid = (NSEG, npairs); deterministic two-pass reduction (no float atomics).
// ---------------------------------------------------------------------------
__global__ __launch_bounds__(256) void dot_partials_kernel(
    const float* __restrict__ Xh, const float* __restrict__ Fh,
    float* __restrict__ partials, int nvalid, int n, int seglen)
{
    __shared__ float red[256];
    // Decode pair index p -> (i, j), i <= j < nvalid.
    int p = blockIdx.y;
    int i = 0, rem = p;
    while (rem >= nvalid - i) { rem -= (nvalid - i); ++i; }
    int j = i + rem;

    const float* xi = Xh + (size_t)i * n;
    const float* fi = Fh + (size_t)i * n;
    const float* xj = Xh + (size_t)j * n;
    const float* fj = Fh + (size_t)j * n;

    const int base = blockIdx.x * seglen;
    float s = 0.0f;
    for (int e = base + threadIdx.x; e < base + seglen; e += 256) {
        float gi = fi[e] - xi[e];
        float gj = fj[e] - xj[e];
        s = fmaf(gi, gj, s);
    }
    red[threadIdx.x] = s;
    __syncthreads();
    for (int off = 128; off > 0; off >>= 1) {
        if (threadIdx.x < off) red[threadIdx.x] += red[threadIdx.x + off];
        __syncthreads();
    }
    if (threadIdx.x == 0) partials[p * gridDim.x + blockIdx.x] = red[0];
}

// ---------------------------------------------------------------------------
// Build bordered system A (11x11) and solve with partial-pivot Gaussian
// elimination; alpha_i = 0 automatically for invalid rows (lam*alpha_i = 0).
// ---------------------------------------------------------------------------
__global__ void solve_kernel(const float* __restrict__ partials,
                             float* __restrict__ alpha, int nvalid, int nseg)
{
    if (threadIdx.x != 0 || blockIdx.x != 0) return;
    const float lam = 1e-4f;
    float H[M_HIST][M_HIST];
    for (int i = 0; i < M_HIST; ++i)
        for (int j = 0; j < M_HIST; ++j) H[i][j] = 0.0f;

    int p = 0;
    for (int i = 0; i < nvalid; ++i)
        for (int j = i; j < nvalid; ++j, ++p) {
            float s = 0.0f;
            for (int sg = 0; sg < nseg; ++sg) s += partials[p * nseg + sg];
            H[i][j] = s;
            H[j][i] = s;
        }
    for (int i = 0; i < M_HIST; ++i) H[i][i] += lam;

    // Augmented [A | rhs], size 11 x 12.
    float A[M_HIST + 1][M_HIST + 2];
    for (int r = 0; r <= M_HIST; ++r)
        for (int c = 0; c <= M_HIST + 1; ++c) A[r][c] = 0.0f;
    for (int i = 0; i < M_HIST; ++i) {
        float mask = (i < nvalid) ? 1.0f : 0.0f;
        A[0][1 + i] = mask;
        A[1 + i][0] = mask;
        for (int j = 0; j < M_HIST; ++j) A[1 + i][1 + j] = H[i][j];
    }
    A[0][M_HIST + 1] = 1.0f;  // rhs[0] = 1

    const int NN = M_HIST + 1;
    for (int col = 0; col < NN; ++col) {
        int piv = col;
        float best = fabsf(A[col][col]);
        for (int r = col + 1; r < NN; ++r) {
            float v = fabsf(A[r][col]);
            if (v > best) { best = v; piv = r; }
        }
        if (piv != col)
            for (int c = col; c <= NN; ++c) {
                float t = A[col][c]; A[col][c] = A[piv][c]; A[piv][c] = t;
            }
        float inv = 1.0f / A[col][col];
        for (int r = col + 1; r < NN; ++r) {
            float fmul = A[r][col] * inv;
            for (int c = col; c <= NN; ++c) A[r][c] -= fmul * A[col][c];
        }
    }
    float xv[M_HIST + 1];
    for (int r = NN - 1; r >= 0; --r) {
        float s = A[r][NN];
        for (int c = r + 1; c < NN; ++c) s -= A[r][c] * xv[c];
        xv[r] = s / A[r][r];
    }
    for (int i = 0; i < M_HIST; ++i) alpha[i] = xv[1 + i];
}

// xk = alpha @ F (beta = 1.0)
__global__ __launch_bounds__(256) void combine_kernel(
    const float* __restrict__ Fh, const float* __restrict__ alpha,
    float* __restrict__ xout, int n)
{
    int e = blockIdx.x * 256 + threadIdx.x;
    if (e >= n) return;
    float s = 0.0f;
#pragma unroll
    for (int i = 0; i < M_HIST; ++i) s = fmaf(alpha[i], Fh[(size_t)i * n + e], s);
    xout[e] = s;
}

__global__ __launch_bounds__(256) void copy_kernel(float* __restrict__ dst,
                                                   const float* __restrict__ src, int n)
{
    int e = blockIdx.x * 256 + threadIdx.x;
    if (e < n) dst[e] = src[e];
}

__global__ __launch_bounds__(256) void zero_kernel(float* __restrict__ dst, int n)
{
    int e = blockIdx.x * 256 + threadIdx.x;
    if (e < n) dst[e] = 0.0f;
}

// ---------------------------------------------------------------------------
extern "C" void kernel_launch(void* const* d_in, const int* in_sizes, int n_in,
                              void* d_out, int out_size, void* d_ws, size_t ws_size,
                              hipStream_t stream)
{
    const float* x   = (const float*)d_in[0];  // [256,1024]
    const float* W   = (const float*)d_in[1];  // [1024,1024]
    const float* b   = (const float*)d_in[2];  // [1024]
    const float* inj = (const float*)d_in[3];  // [256,1024]
    float* out = (float*)d_out;

    const int n = N_FLAT;
    float* Xh = (float*)d_ws;                       // 10 rows of n
    float* Fh = Xh + (size_t)M_HIST * n;            // 10 rows of n
    float* partials = Fh + (size_t)M_HIST * n;      // 55 * NSEG
    float* alpha = partials + MAXPAIR * NSEG;       // 10

    const dim3 gemmGrid(D_DIM / 64, BATCH / 128);   // (16, 2)
    const int nBlk = n / 256;

    // X[0] = x0 ; F[0] = f(x0) ; X[1] = f0 ; F[1] = f(f0) ; rows 2..9 zeroed.
    copy_kernel<<<nBlk, 256, 0, stream>>>(Xh, x, n);
    f_apply_kernel<<<gemmGrid, 256, 0, stream>>>(x, W, inj, b, Fh);
    copy_kernel<<<nBlk, 256, 0, stream>>>(Xh + (size_t)n, Fh, n);
    f_apply_kernel<<<gemmGrid, 256, 0, stream>>>(Xh + (size_t)n, W, inj, b, Fh + (size_t)n);
    zero_kernel<<<(8 * n) / 256, 256, 0, stream>>>(Xh + (size_t)2 * n, 8 * n);
    zero_kernel<<<(8 * n) / 256, 256, 0, stream>>>(Fh + (size_t)2 * n, 8 * n);

    for (int k = 2; k < 100; ++k) {
        const int nvalid = (k < M_HIST) ? k : M_HIST;
        const int npairs = nvalid * (nvalid + 1) / 2;
        dot_partials_kernel<<<dim3(NSEG, npairs), 256, 0, stream>>>(
            Xh, Fh, partials, nvalid, n, n / NSEG);
        solve_kernel<<<1, 1, 0, stream>>>(partials, alpha, nvalid, NSEG);
        const int idx = k % M_HIST;
        combine_kernel<<<nBlk, 256, 0, stream>>>(Fh, alpha, Xh + (size_t)idx * n, n);
        f_apply_kernel<<<gemmGrid, 256, 0, stream>>>(
            Xh + (size_t)idx * n, W, inj, b, Fh + (size_t)idx * n);
    }

    // x_star = xk of last step (k=99 -> slot 9).
    copy_kernel<<<nBlk, 256, 0, stream>>>(out, Xh + (size_t)9 * n, n);
}